// LinearCondensed_17016660427310
// MI455X (gfx1250) — compile-verified
//
#include <hip/hip_runtime.h>

// LinearCondensed: out[b,o] = sum_f weight[o,f] * input[b, idx[o,f]] + bias[o]
// B=512, IN_WIDTH=4096, O=4096, FAN_IN=128, all f32 (idx passed as int32 per harness contract).

typedef __attribute__((ext_vector_type(2))) float v2f;
typedef __attribute__((ext_vector_type(8))) float v8f;

#define BATCH        512
#define IN_WIDTH     4096
#define OUT_FEATURES 4096
#define FAN_IN       128

#define BTILE        16
// 4096 + 4 pad floats: row stride == 4 (mod 64 banks) -> a 16-row column gather
// (addresses m*4100 + col, m=0..15) touches 16 distinct LDS banks. Staging stays
// contiguous/aligned for ds_store_b128.
#define LDS_STRIDE   4100
#define THREADS      256
#define WAVES        (THREADS / 32)
#define FEATS_PER_WG 512
#define FTILES_PER_WAVE (FEATS_PER_WG / (16 * WAVES))   // = 4

__global__ __launch_bounds__(THREADS, 1)
void lincond_wmma_f32(const float* __restrict__ input,
                      const float* __restrict__ weight,
                      const float* __restrict__ bias,
                      const int*   __restrict__ indx,
                      float*       __restrict__ out)
{
    // 16 x 4100 f32 = 262,400 bytes (WGP LDS is 320 KB on CDNA5)
    __shared__ float lds[BTILE * LDS_STRIDE];

    const int btile  = blockIdx.x;        // which 16-row batch tile
    const int fchunk = blockIdx.y;        // which 512-feature chunk
    const int t      = threadIdx.x;

    // ---------------- Stage input tile (16 x 4096) into padded LDS ----------------
    // Coalesced float4 global loads, 16B-aligned LDS stores (4100 % 4 == 0).
    {
        const float4* in4 = (const float4*)(input + (size_t)btile * BTILE * IN_WIDTH);
        #pragma unroll
        for (int i = 0; i < (BTILE * IN_WIDTH / 4) / THREADS; ++i) {
            const int e  = t + i * THREADS;          // 0 .. 16383
            const int r  = e >> 10;                  // row 0..15   (1024 float4 per row)
            const int c4 = e & 1023;                 // float4 column
            const float4 v = in4[(size_t)r * (IN_WIDTH / 4) + c4];
            *(float4*)&lds[r * LDS_STRIDE + c4 * 4] = v;
        }
    }
    __syncthreads();

    // ---------------- WMMA compute: one 16x16 (batch x ofeat) tile at a time -------
    // A-matrix f32 16x4 layout: lanes 0-15 hold (M=lane, K=0/1), lanes 16-31 (M=lane-16, K=2/3).
    // The two K columns a lane needs are CONSECUTIVE fan-in slots -> per-lane b64 vector
    // loads of idx/weight pairs land directly in the right lanes (no selects, no SGPR use).
    const int  lane  = t & 31;
    const int  wave  = t >> 5;
    const bool lo    = (lane < 16);
    const int  m     = lane & 15;          // A-matrix row (batch within tile)
    const int  ncol  = lane & 15;          // B/C/D column (ofeat within tile)
    const int  mbase = m * LDS_STRIDE;     // lane's LDS row base (element units)
    const int  kbase = lo ? 0 : 2;         // which K-pair this lane supplies

    for (int ft = 0; ft < FTILES_PER_WAVE; ++ft) {
        const int o_tile = __builtin_amdgcn_readfirstlane(
            fchunk * FEATS_PER_WG + (wave * FTILES_PER_WAVE + ft) * 16);

        v8f acc0 = {};   // two independent WMMA chains to cover XDL latency
        v8f acc1 = {};

        for (int n = 0; n < 16; ++n) {     // block-diagonal: feature n owns B column n
            const int o = o_tile + n;
            // per-lane base pointers; +kbase keeps 8B alignment (o*128 is 512B aligned)
            const int*   ip = indx   + (size_t)o * FAN_IN + kbase;
            const float* wp = weight + (size_t)o * FAN_IN + kbase;
            const bool  onn = (ncol == n);   // loop-invariant VCC for B masking

            #pragma unroll
            for (int fc = 0; fc < FAN_IN; fc += 8) {
                {   // fan-in elements [fc, fc+4)
                    const int2   iv = *(const int2*)(ip + fc);    // cols k, k+1 for this lane
                    const float2 wv = *(const float2*)(wp + fc);
                    v2f a, b;
                    a.x = lds[mbase + iv.x];
                    a.y = lds[mbase + iv.y];
                    b.x = onn ? wv.x : 0.0f;
                    b.y = onn ? wv.y : 0.0f;
                    acc0 = __builtin_amdgcn_wmma_f32_16x16x4_f32(
                        false, a, false, b, (short)0, acc0, false, false);
                }
                {   // fan-in elements [fc+4, fc+8)
                    const int2   iv = *(const int2*)(ip + fc + 4);
                    const float2 wv = *(const float2*)(wp + fc + 4);
                    v2f a, b;
                    a.x = lds[mbase + iv.x];
                    a.y = lds[mbase + iv.y];
                    b.x = onn ? wv.x : 0.0f;
                    b.y = onn ? wv.y : 0.0f;
                    acc1 = __builtin_amdgcn_wmma_f32_16x16x4_f32(
                        false, a, false, b, (short)0, acc1, false, false);
                }
            }
        }

        // ---------------- Epilogue: bias + store -------------------------------
        // C/D layout: VGPR j -> lanes 0-15 row M=j, lanes 16-31 row M=j+8; N = lane&15.
        const float  bv   = bias[o_tile + ncol];
        const size_t row0 = (size_t)btile * BTILE + (lo ? 0 : 8);
        float* op = out + row0 * OUT_FEATURES + o_tile + ncol;
        #pragma unroll
        for (int j = 0; j < 8; ++j) {
            op[(size_t)j * OUT_FEATURES] = acc0[j] + acc1[j] + bv;
        }
    }
}

extern "C" void kernel_launch(void* const* d_in, const int* in_sizes, int n_in,
                              void* d_out, int out_size, void* d_ws, size_t ws_size,
                              hipStream_t stream) {
    (void)in_sizes; (void)n_in; (void)d_ws; (void)ws_size; (void)out_size;
    const float* input  = (const float*)d_in[0];
    const float* weight = (const float*)d_in[1];
    const float* bias   = (const float*)d_in[2];
    const int*   indx   = (const int*)d_in[3];
    float* out = (float*)d_out;

    dim3 grid(BATCH / BTILE, OUT_FEATURES / FEATS_PER_WG);   // 32 x 8
    lincond_wmma_f32<<<grid, THREADS, 0, stream>>>(input, weight, bias, indx, out);
}